// GATv2Rel_83330955477482
// MI455X (gfx1250) — compile-verified
//
#include <hip/hip_runtime.h>
#include <math.h>

typedef float v2f __attribute__((ext_vector_type(2)));
typedef float v8f __attribute__((ext_vector_type(8)));

#define NN 50000
#define FF 300
#define DD 128
#define HH 4
#define CC 32
#define EE 640000
#define RR 200
#define LL 2

// ---------------------------------------------------------------------------
// float atomic max via sign-split int/uint atomics (works with -inf init)
// ---------------------------------------------------------------------------
__device__ __forceinline__ void atomicMaxFloat(float* addr, float val) {
  if (val >= 0.0f) {
    atomicMax((int*)addr, __float_as_int(val));
  } else {
    atomicMin((unsigned int*)addr, __float_as_uint(val));
  }
}

// ---------------------------------------------------------------------------
// Dense GEMM:  Cout[M x 128] = A[M x K] (optionally row-gathered) @ B[K x 128] + bias
// One block = one 16-row tile; 8 waves/block, wave w owns 16-col tile w.
// fp32 WMMA 16x16x4, ISA VGPR layouts:
//   A: lane(l<16): K=k,k+1 ; lane(l>=16): K=k+2,k+3 ; M = lane&15
//   B: lane&15 = N col; same K striping
//   C/D: vgpr v -> M = v + (lane>=16 ? 8 : 0); N = lane&15
// ---------------------------------------------------------------------------
__global__ __launch_bounds__(256) void gemm_f32_wmma(
    const float* __restrict__ A, const float* __restrict__ B,
    const float* __restrict__ bias, float* __restrict__ Cout,
    const int* __restrict__ rowidx, int K, int lda)
{
  const int wave = threadIdx.x >> 5;
  const int lane = threadIdx.x & 31;
  const int half = lane >> 4;       // 0: lanes 0-15, 1: lanes 16-31
  const int m15  = lane & 15;
  const int row0 = blockIdx.x * 16;
  const int col0 = wave * 16;

  int arow = row0 + m15;
  if (rowidx) arow = rowidx[arow];  // embedding-lookup fused into A row gather
  const float* __restrict__ Ar = A + (size_t)arow * (size_t)lda;

  v8f acc = {};
  for (int k = 0; k < K; k += 4) {
    const int kk = k + half * 2;
    v2f a, b;
    a.x = Ar[kk];
    a.y = Ar[kk + 1];
    b.x = B[(size_t)kk       * DD + col0 + m15];
    b.y = B[(size_t)(kk + 1) * DD + col0 + m15];
    acc = __builtin_amdgcn_wmma_f32_16x16x4_f32(
        /*neg_a=*/false, a, /*neg_b=*/false, b,
        /*c_mod=*/(short)0, acc, /*reuse_a=*/false, /*reuse_b=*/false);
  }

  const float bv = bias ? bias[col0 + m15] : 0.0f;
#pragma unroll
  for (int v = 0; v < 8; ++v) {
    const int row = row0 + half * 8 + v;
    Cout[(size_t)row * DD + col0 + m15] = acc[v] + bv;
  }
}

// ---------------------------------------------------------------------------
// e_rel[200 x 128] = rel_emb[200 x 128] @ w_e[128 x 128]   (tiny, scalar)
// ---------------------------------------------------------------------------
__global__ __launch_bounds__(256) void erel_gemm_k(
    const float* __restrict__ rel, const float* __restrict__ we,
    float* __restrict__ erel)
{
  const int i = blockIdx.x * 256 + threadIdx.x;
  if (i >= RR * DD) return;
  const int r = i >> 7, n = i & 127;
  float s = 0.0f;
  for (int k = 0; k < DD; ++k) s += rel[r * DD + k] * we[k * DD + n];
  erel[i] = s;
}

__global__ __launch_bounds__(256) void fill_k(float* __restrict__ p, float v, long n) {
  const long i = (long)blockIdx.x * 256 + threadIdx.x;
  if (i < n) p[i] = v;
}

// ---------------------------------------------------------------------------
// Pass 1: per-(edge,head) attention logit + segment max.
// One wave per (edge,head); lane = channel -> coalesced 128B loads.
// ---------------------------------------------------------------------------
__global__ __launch_bounds__(256) void edge_logits_k(
    const float* __restrict__ xl, const float* __restrict__ xr,
    const float* __restrict__ erel, const int* __restrict__ ei,
    const int* __restrict__ etype, const float* __restrict__ att,
    float* __restrict__ logits, float* __restrict__ lmax)
{
  const int wave = threadIdx.x >> 5, lane = threadIdx.x & 31;
  const long eh = (long)blockIdx.x * 8 + wave;
  if (eh >= (long)EE * HH) return;
  const int e = (int)(eh >> 2), h = (int)(eh & 3);
  const int src = ei[e], dst = ei[EE + e], t = etype[e];
  const int c = h * CC + lane;

  float v = xl[(size_t)src * DD + c] + xr[(size_t)dst * DD + c]
          + erel[(size_t)t * DD + c];
  v = v > 0.0f ? v : 0.2f * v;          // GATv2: leaky_relu BEFORE att dot
  v *= att[c];
  for (int off = 16; off; off >>= 1) v += __shfl_xor(v, off, 32);
  if (lane == 0) {
    logits[eh] = v;
    atomicMaxFloat(&lmax[(size_t)dst * HH + h], v);
  }
}

// ---------------------------------------------------------------------------
// Pass 2: ex = exp(logit - lmax[dst]); denom[dst] += ex   (logits overwritten)
// ---------------------------------------------------------------------------
__global__ __launch_bounds__(256) void edge_exp_k(
    float* __restrict__ logits, const float* __restrict__ lmax,
    float* __restrict__ denom, const int* __restrict__ ei)
{
  const long i = (long)blockIdx.x * 256 + threadIdx.x;
  if (i >= (long)EE * HH) return;
  const int e = (int)(i >> 2), h = (int)(i & 3);
  const int dst = ei[EE + e];
  const float ex = __expf(logits[i] - lmax[(size_t)dst * HH + h]);
  logits[i] = ex;
  atomicAdd(&denom[(size_t)dst * HH + h], ex);
}

// ---------------------------------------------------------------------------
// Pass 3: xagg[dst] += alpha * x_l[src]; wave per (edge,head), coalesced atomics
// ---------------------------------------------------------------------------
__global__ __launch_bounds__(256) void edge_scatter_k(
    const float* __restrict__ xl, const float* __restrict__ exbuf,
    const float* __restrict__ denom, const int* __restrict__ ei,
    float* __restrict__ xagg)
{
  const int wave = threadIdx.x >> 5, lane = threadIdx.x & 31;
  const long eh = (long)blockIdx.x * 8 + wave;
  if (eh >= (long)EE * HH) return;
  const int e = (int)(eh >> 2), h = (int)(eh & 3);
  const int src = ei[e], dst = ei[EE + e];
  const float alpha = exbuf[eh] / denom[(size_t)dst * HH + h];
  const int c = h * CC + lane;
  atomicAdd(&xagg[(size_t)dst * DD + c], xl[(size_t)src * DD + c] * alpha);
}

// ---------------------------------------------------------------------------
// out = xagg + bias, optional ELU (between layers)
// ---------------------------------------------------------------------------
__global__ __launch_bounds__(256) void finalize_k(
    const float* __restrict__ xagg, const float* __restrict__ bias,
    float* __restrict__ dst, int apply_elu)
{
  const long i = (long)blockIdx.x * 256 + threadIdx.x;
  if (i >= (long)NN * DD) return;
  float v = xagg[i] + bias[i & (DD - 1)];
  if (apply_elu) v = v > 0.0f ? v : expm1f(v);
  dst[i] = v;
}

// ---------------------------------------------------------------------------
extern "C" void kernel_launch(void* const* d_in, const int* in_sizes, int n_in,
                              void* d_out, int out_size, void* d_ws, size_t ws_size,
                              hipStream_t stream)
{
  (void)in_sizes; (void)n_in; (void)out_size; (void)ws_size;

  const int*   entity        = (const int*)d_in[0];
  const int*   edge_index    = (const int*)d_in[1];
  const int*   edge_type     = (const int*)d_in[2];
  const float* node_features = (const float*)d_in[3];
  const float* rel_emb       = (const float*)d_in[4];
  const float* proj_w        = (const float*)d_in[5];
  const float* proj_b        = (const float*)d_in[6];
  const float* w_l           = (const float*)d_in[7];
  const float* b_l           = (const float*)d_in[8];
  const float* w_r           = (const float*)d_in[9];
  const float* b_r           = (const float*)d_in[10];
  const float* w_e           = (const float*)d_in[11];
  const float* att           = (const float*)d_in[12];
  const float* bias          = (const float*)d_in[13];
  float*       out           = (float*)d_out;

  // workspace layout (floats)
  float* x      = (float*)d_ws;
  float* xl     = x      + (size_t)NN * DD;
  float* xr     = xl     + (size_t)NN * DD;
  float* xagg   = xr     + (size_t)NN * DD;
  float* erel   = xagg   + (size_t)NN * DD;
  float* logits = erel   + (size_t)RR * DD;
  float* lmax   = logits + (size_t)EE * HH;
  float* denom  = lmax   + (size_t)NN * HH;

  const long nh = (long)NN * HH;
  const long nd = (long)NN * DD;
  const long ehn = (long)EE * HH;

  // x = node_features[entity] @ proj_w + proj_b   (WMMA fp32)
  gemm_f32_wmma<<<NN / 16, 256, 0, stream>>>(node_features, proj_w, proj_b, x,
                                             entity, FF, FF);

  for (int l = 0; l < LL; ++l) {
    gemm_f32_wmma<<<NN / 16, 256, 0, stream>>>(x, w_l + (size_t)l * DD * DD,
                                               b_l + l * DD, xl, nullptr, DD, DD);
    gemm_f32_wmma<<<NN / 16, 256, 0, stream>>>(x, w_r + (size_t)l * DD * DD,
                                               b_r + l * DD, xr, nullptr, DD, DD);
    // factorization: e[edge] = (rel_emb @ w_e)[edge_type[edge]]
    erel_gemm_k<<<(RR * DD + 255) / 256, 256, 0, stream>>>(
        rel_emb, w_e + (size_t)l * DD * DD, erel);

    fill_k<<<(int)((nh + 255) / 256), 256, 0, stream>>>(lmax, -INFINITY, nh);
    fill_k<<<(int)((nh + 255) / 256), 256, 0, stream>>>(denom, 0.0f, nh);
    fill_k<<<(int)((nd + 255) / 256), 256, 0, stream>>>(xagg, 0.0f, nd);

    edge_logits_k<<<(int)(ehn / 8), 256, 0, stream>>>(
        xl, xr, erel, edge_index, edge_type, att + l * HH * CC, logits, lmax);
    edge_exp_k<<<(int)((ehn + 255) / 256), 256, 0, stream>>>(
        logits, lmax, denom, edge_index);
    edge_scatter_k<<<(int)(ehn / 8), 256, 0, stream>>>(
        xl, logits, denom, edge_index, xagg);

    finalize_k<<<(int)((nd + 255) / 256), 256, 0, stream>>>(
        xagg, bias + l * DD, (l == LL - 1) ? out : x, (l != LL - 1) ? 1 : 0);
  }
}